// GNN_5970004542313
// MI455X (gfx1250) — compile-verified
//
#include <hip/hip_runtime.h>
#include <math.h>

typedef __attribute__((ext_vector_type(2))) float v2f;
typedef __attribute__((ext_vector_type(8))) float v8f;

// ---------------- edge / node kernels (memory-bound part) ----------------

__global__ void k_init_deg(float* __restrict__ deg, int n) {
  int i = blockIdx.x * blockDim.x + threadIdx.x;
  if (i < n) deg[i] = 1.0f;  // self-loop contributes 1 to every target degree
}

__global__ void k_degree(const int* __restrict__ dst, float* __restrict__ deg, int e) {
  int i = blockIdx.x * blockDim.x + threadIdx.x;
  if (i < e) atomicAdd(&deg[dst[i]], 1.0f);
}

// dinv = rsqrt(deg) (in place), agg1 seeded with self-loop term dinv^2 * x
__global__ void k_dinv_seed(const float* __restrict__ x, float* __restrict__ deg_dinv,
                            float* __restrict__ agg1, int n) {
  int i = blockIdx.x * blockDim.x + threadIdx.x;
  if (i < n) {
    float di = rsqrtf(deg_dinv[i]);  // deg >= 1 always (self-loop)
    deg_dinv[i] = di;
    agg1[i] = di * di * x[i];
  }
}

// layer-1 edge aggregation: scalar per edge (Fin == 1)
__global__ void k_l1_edges(const int* __restrict__ src, const int* __restrict__ dst,
                           const float* __restrict__ x, const float* __restrict__ dinv,
                           float* __restrict__ agg1, int e) {
  int i = blockIdx.x * blockDim.x + threadIdx.x;
  if (i < e) {
    int s = src[i], d = dst[i];
    atomicAdd(&agg1[d], dinv[s] * dinv[d] * x[s]);
  }
}

// ---------------- WMMA node transform ----------------
// Per wave: 16 nodes. A[m][k] = relu(agg1[m]*W1[k]+b1[k])  (16x16, f32)
// B[k][c]  = W2[k][c] for c<2 else 0                       (16x16 padded)
// D = A x B via 4 chained V_WMMA_F32_16X16X4_F32 (K=4 each).
// Outputs h2[N,2] and agg2 seeded with self-loop term dinv^2 * h2.
__global__ void k_node_wmma(const float* __restrict__ agg1, const float* __restrict__ dinv,
                            const float* __restrict__ W1, const float* __restrict__ b1,
                            const float* __restrict__ W2,
                            float* __restrict__ h2, float* __restrict__ agg2, int n) {
  const int lane = threadIdx.x & 31;
  const int wave = threadIdx.x >> 5;
  const int gw   = blockIdx.x * (blockDim.x >> 5) + wave;
  const int nodeBase = gw * 16;

  const int m    = lane & 15;   // A: M row / B: N column for this lane
  const int half = lane >> 4;   // lane half selects K sub-rows

  int node = nodeBase + m;
  if (node >= n) node = n - 1;       // clamp loads; keep EXEC all-ones for WMMA
  const float a1 = agg1[node];

  v8f c = {0.f, 0.f, 0.f, 0.f, 0.f, 0.f, 0.f, 0.f};
#pragma unroll
  for (int s = 0; s < 4; ++s) {
    const int k0 = 4 * s + 2 * half;  // this lane's two K indices for step s
    v2f a, b;
    // A 16x4 f32 layout: lanes 0-15 M=0..15 hold K = 4s+0,4s+1; lanes 16-31 K = 4s+2,4s+3
    a.x = fmaxf(0.f, fmaf(a1, W1[k0],     b1[k0]));
    a.y = fmaxf(0.f, fmaf(a1, W1[k0 + 1], b1[k0 + 1]));
    // B 4x16 f32 layout: lane's column = m; rows mirror A's K split; pad cols >= 2 with 0
    b.x = (m < 2) ? W2[k0 * 2 + m]       : 0.f;
    b.y = (m < 2) ? W2[(k0 + 1) * 2 + m] : 0.f;
    c = __builtin_amdgcn_wmma_f32_16x16x4_f32(false, a, false, b, (short)0, c, false, false);
  }

  // D 16x16 f32 layout: lane column = m; VGPR v holds row (v + 8*half).
  // Only columns 0,1 are real outputs (lanes 0,1,16,17).
  if (m < 2) {
#pragma unroll
    for (int v = 0; v < 8; ++v) {
      const int node2 = nodeBase + half * 8 + v;
      if (node2 < n) {
        const float val = c[v];
        const float di  = dinv[node2];
        h2[node2 * 2 + m]   = val;
        agg2[node2 * 2 + m] = di * di * val;  // self-loop term of layer 2
      }
    }
  }
}

// layer-2 edge aggregation: 2 floats per edge (W2 applied before scatter)
__global__ void k_l2_edges(const int* __restrict__ src, const int* __restrict__ dst,
                           const float* __restrict__ h2, const float* __restrict__ dinv,
                           float* __restrict__ agg2, int e) {
  int i = blockIdx.x * blockDim.x + threadIdx.x;
  if (i < e) {
    int s = src[i], d = dst[i];
    float nm = dinv[s] * dinv[d];
    atomicAdd(&agg2[2 * d + 0], nm * h2[2 * s + 0]);
    atomicAdd(&agg2[2 * d + 1], nm * h2[2 * s + 1]);
  }
}

// bias + 2-class log_softmax
__global__ void k_logsoftmax(const float* __restrict__ agg2, const float* __restrict__ b2,
                             float* __restrict__ out, int n) {
  int i = blockIdx.x * blockDim.x + threadIdx.x;
  if (i < n) {
    float o0 = agg2[2 * i + 0] + b2[0];
    float o1 = agg2[2 * i + 1] + b2[1];
    float mx = fmaxf(o0, o1);
    float lse = mx + logf(expf(o0 - mx) + expf(o1 - mx));
    out[2 * i + 0] = o0 - lse;
    out[2 * i + 1] = o1 - lse;
  }
}

// ---------------- launch ----------------

extern "C" void kernel_launch(void* const* d_in, const int* in_sizes, int n_in,
                              void* d_out, int out_size, void* d_ws, size_t ws_size,
                              hipStream_t stream) {
  const float* x   = (const float*)d_in[0];
  const int*   ei  = (const int*)d_in[1];
  const float* W1  = (const float*)d_in[2];
  const float* b1  = (const float*)d_in[3];
  const float* W2  = (const float*)d_in[4];
  const float* b2  = (const float*)d_in[5];
  float* out = (float*)d_out;

  const int N = in_sizes[0];      // 100000
  const int E = in_sizes[1] / 2;  // 6400000
  const int* src = ei;            // edge_index[0]
  const int* dst = ei + E;        // edge_index[1]

  // workspace carve-up (floats): dinv[N] | agg1[N] | h2[2N] | agg2[2N]
  float* dinv = (float*)d_ws;     // starts life as deg, overwritten in-place by rsqrt
  float* agg1 = dinv + N;
  float* h2   = agg1 + N;
  float* agg2 = h2 + 2 * N;

  const int TB = 256;
  const int gN = (N + TB - 1) / TB;
  const int gE = (E + TB - 1) / TB;
  // node WMMA kernel: 16 nodes per wave, 8 waves per 256-thread block
  const int waves  = (N + 15) / 16;
  const int gWmma  = (waves + 7) / 8;

  k_init_deg   <<<gN,    TB, 0, stream>>>(dinv, N);
  k_degree     <<<gE,    TB, 0, stream>>>(dst, dinv, E);
  k_dinv_seed  <<<gN,    TB, 0, stream>>>(x, dinv, agg1, N);
  k_l1_edges   <<<gE,    TB, 0, stream>>>(src, dst, x, dinv, agg1, E);
  k_node_wmma  <<<gWmma, TB, 0, stream>>>(agg1, dinv, W1, b1, W2, h2, agg2, N);
  k_l2_edges   <<<gE,    TB, 0, stream>>>(src, dst, h2, dinv, agg2, E);
  k_logsoftmax <<<gN,    TB, 0, stream>>>(agg2, b2, out, N);
}